// TFEncoder_80049600463012
// MI455X (gfx1250) — compile-verified
//
#include <hip/hip_runtime.h>
#include <math.h>

typedef __attribute__((ext_vector_type(2))) float v2f;
typedef __attribute__((ext_vector_type(8))) float v8f;

#define WMMA_F32(a, b, c) \
  __builtin_amdgcn_wmma_f32_16x16x4_f32(false, (a), false, (b), (short)0, (c), false, false)

// ---- CDNA5 async global->LDS staging (ASYNCcnt DMA), with sync fallback ----
#if defined(__has_builtin)
#if __has_builtin(__builtin_amdgcn_global_load_async_to_lds_b128)
#define USE_ASYNC_LDS 1
#endif
#endif
#ifndef USE_ASYNC_LDS
#define USE_ASYNC_LDS 0
#endif

#if USE_ASYNC_LDS
#define AS1 __attribute__((address_space(1)))
#define AS3 __attribute__((address_space(3)))
typedef int v4i_b128 __attribute__((vector_size(16)));
static __device__ __forceinline__ void async_cp16(float* lds, const float* g) {
  // copies 16 bytes from the global address g to the LDS address lds, tracked by ASYNCcnt
  __builtin_amdgcn_global_load_async_to_lds_b128(
      (AS1 v4i_b128*)(g), (AS3 v4i_b128*)(lds), 0, 0);
}
#if __has_builtin(__builtin_amdgcn_s_wait_asynccnt)
#define WAIT_ASYNC() __builtin_amdgcn_s_wait_asynccnt(0)
#else
#define WAIT_ASYNC() asm volatile("s_wait_asynccnt 0x0" ::: "memory")
#endif
#else
#define WAIT_ASYNC()
#endif

static __device__ __forceinline__ float redmax16(float v) {
  v = fmaxf(v, __shfl_xor(v, 1, 32));
  v = fmaxf(v, __shfl_xor(v, 2, 32));
  v = fmaxf(v, __shfl_xor(v, 4, 32));
  v = fmaxf(v, __shfl_xor(v, 8, 32));
  return v;
}
static __device__ __forceinline__ float redsum16(float v) {
  v += __shfl_xor(v, 1, 32);
  v += __shfl_xor(v, 2, 32);
  v += __shfl_xor(v, 4, 32);
  v += __shfl_xor(v, 8, 32);
  return v;
}

// ---------------------------------------------------------------------------
// fp32 WMMA GEMM: C[z] = A(MxK) * B[z](KxN) (+bias)(+relu)
// Block = 128 threads (4 waves), tile 128(M) x 64(N), BK = 32, double-buffered
// LDS with async global->LDS staging. Each wave: 32 rows -> 8 WMMA / k-step.
// Requires M%128==0, N%64==0, K%32==0 (true for all uses here).
// ---------------------------------------------------------------------------
__global__ __launch_bounds__(128)
void gemm_wmma(const float* __restrict__ A, int lda,
               const float* __restrict__ B, int ldb, long long bStrideZ,
               float* __restrict__ C, int ldc, long long cStrideZ,
               int K, const float* __restrict__ bias, int flags)
{
  __shared__ __align__(16) float lA[2][128][36];  // pad: conflict-free, 16B-aligned
  __shared__ __align__(16) float lB[2][32][80];
  const int tid  = threadIdx.x;
  const int wv   = tid >> 5;
  const int lane = tid & 31;
  const int hl   = lane >> 4;
  const int l16  = lane & 15;
  const int m0   = blockIdx.x * 128;
  const int n0   = blockIdx.y * 64;
  const float* Bz = B + (long long)blockIdx.z * bStrideZ;
  float*       Cz = C + (long long)blockIdx.z * cStrideZ;

  const v8f zero8 = {0, 0, 0, 0, 0, 0, 0, 0};
  v8f acc[2][4];
#pragma unroll
  for (int mt = 0; mt < 2; ++mt)
#pragma unroll
    for (int nt = 0; nt < 4; ++nt) acc[mt][nt] = zero8;

  auto stage = [&](int buf, int k0) {
#pragma unroll
    for (int i = 0; i < 8; ++i) {          // A tile 128x32
      int idx = (tid + i * 128) * 4;
      int r = idx >> 5, c = idx & 31;
      const float* src = A + (size_t)(m0 + r) * lda + k0 + c;
#if USE_ASYNC_LDS
      async_cp16(&lA[buf][r][c], src);
#else
      *(float4*)&lA[buf][r][c] = *(const float4*)src;
#endif
    }
#pragma unroll
    for (int i = 0; i < 4; ++i) {          // B tile 32x64
      int idx = (tid + i * 128) * 4;
      int r = idx >> 6, c = idx & 63;
      const float* src = Bz + (size_t)(k0 + r) * ldb + n0 + c;
#if USE_ASYNC_LDS
      async_cp16(&lB[buf][r][c], src);
#else
      *(float4*)&lB[buf][r][c] = *(const float4*)src;
#endif
    }
  };

  stage(0, 0);
  int cur = 0;
  for (int k0 = 0; k0 < K; k0 += 32) {
    WAIT_ASYNC();          // my staged copies have landed in LDS
    __syncthreads();       // everyone's have; previous compute fully drained
    if (k0 + 32 < K) stage(cur ^ 1, k0 + 32);   // DMA next slab during compute
#pragma unroll
    for (int kk = 0; kk < 32; kk += 4) {
      v2f a0, a1;
      a0.x = lA[cur][wv * 32 + l16][kk + 2 * hl];
      a0.y = lA[cur][wv * 32 + l16][kk + 2 * hl + 1];
      a1.x = lA[cur][wv * 32 + 16 + l16][kk + 2 * hl];
      a1.y = lA[cur][wv * 32 + 16 + l16][kk + 2 * hl + 1];
#pragma unroll
      for (int nt = 0; nt < 4; ++nt) {
        v2f b;
        b.x = lB[cur][kk + 2 * hl][nt * 16 + l16];
        b.y = lB[cur][kk + 2 * hl + 1][nt * 16 + l16];
        acc[0][nt] = WMMA_F32(a0, b, acc[0][nt]);
        acc[1][nt] = WMMA_F32(a1, b, acc[1][nt]);
      }
    }
    cur ^= 1;
  }
#pragma unroll
  for (int mt = 0; mt < 2; ++mt) {
#pragma unroll
    for (int nt = 0; nt < 4; ++nt) {
#pragma unroll
      for (int j = 0; j < 8; ++j) {
        int row = m0 + wv * 32 + mt * 16 + j + 8 * hl;  // C: M = vgpr + 8*(lane/16)
        int col = n0 + nt * 16 + l16;
        float vv = acc[mt][nt][j];
        if (bias) vv += bias[col];
        if (flags & 1) vv = fmaxf(vv, 0.0f);
        Cz[(size_t)row * ldc + col] = vv;
      }
    }
  }
}

// ---------------------------------------------------------------------------
// Flash attention, fp32 WMMA. q/k/v laid out [H][B*S][64].
// Block = 128 threads (4 waves) handles 64 query rows of one (b,h);
// each wave owns 16 rows. Online softmax, no SxS materialization.
// ---------------------------------------------------------------------------
#define S_LEN 2048
#define ADIM  64

__global__ __launch_bounds__(128)
void attn_flash(const float* __restrict__ Q, const float* __restrict__ Km,
                const float* __restrict__ V, float* __restrict__ Z)
{
  __shared__ __align__(16) float ktT[64][40];     // K transposed: [a][key row]
  __shared__ __align__(16) float vT[32][80];      // V: [key row][a]
  __shared__ __align__(16) float pbuf[4][16][18]; // per-wave P relayout C->A
  const int tid  = threadIdx.x;
  const int wv   = tid >> 5;
  const int lane = tid & 31;
  const int hl   = lane >> 4;
  const int l16  = lane & 15;
  const int qt = blockIdx.x, h = blockIdx.y, b = blockIdx.z;
  const size_t MTOT = (size_t)4 * S_LEN;
  const float* Qh = Q  + ((size_t)h * MTOT + (size_t)b * S_LEN + qt * 64) * ADIM;
  const float* Kh = Km + ((size_t)h * MTOT + (size_t)b * S_LEN) * ADIM;
  const float* Vh = V  + ((size_t)h * MTOT + (size_t)b * S_LEN) * ADIM;

  // preload this wave's 16 query rows as A-fragments, pre-scaled by 1/sqrt(D)
  const float scale = 0.03125f;  // 1/sqrt(1024)
  v2f aQ[16];
#pragma unroll
  for (int kk4 = 0; kk4 < 16; ++kk4) {
    const float* p = Qh + (size_t)(wv * 16 + l16) * ADIM + kk4 * 4 + 2 * hl;
    aQ[kk4].x = p[0] * scale;
    aQ[kk4].y = p[1] * scale;
  }

  float mrun[8], lrun[8];
#pragma unroll
  for (int j = 0; j < 8; ++j) { mrun[j] = -3.0e38f; lrun[j] = 0.0f; }
  const v8f zero8 = {0, 0, 0, 0, 0, 0, 0, 0};
  v8f accO[4] = {zero8, zero8, zero8, zero8};

  for (int kt = 0; kt < S_LEN; kt += 32) {
    __syncthreads();
#pragma unroll
    for (int i = 0; i < 4; ++i) {
      int idx = (tid + i * 128) * 4;       // over 32x64 tile
      int r = idx >> 6, c = idx & 63;
      float4 kv = *(const float4*)(Kh + (size_t)(kt + r) * ADIM + c);
      ktT[c + 0][r] = kv.x; ktT[c + 1][r] = kv.y;   // transpose: must go via VGPRs
      ktT[c + 2][r] = kv.z; ktT[c + 3][r] = kv.w;
      const float* vsrc = Vh + (size_t)(kt + r) * ADIM + c;
#if USE_ASYNC_LDS
      async_cp16(&vT[r][c], vsrc);                  // V: contiguous -> async DMA
#else
      *(float4*)&vT[r][c] = *(const float4*)vsrc;
#endif
    }
    WAIT_ASYNC();
    __syncthreads();
#pragma unroll
    for (int sub = 0; sub < 2; ++sub) {
      // S = (Q*scale) @ K^T for 16 keys: 16 WMMAs over ADIM
      v8f s = zero8;
#pragma unroll
      for (int kk4 = 0; kk4 < 16; ++kk4) {
        v2f bk;
        bk.x = ktT[kk4 * 4 + 2 * hl][sub * 16 + l16];
        bk.y = ktT[kk4 * 4 + 2 * hl + 1][sub * 16 + l16];
        s = WMMA_F32(aQ[kk4], bk, s);
      }
      // online softmax update (rows j + 8*hl; reduce over 16 col-lanes)
      float alpha[8];
#pragma unroll
      for (int j = 0; j < 8; ++j) {
        float tm = redmax16(s[j]);
        float nm = fmaxf(mrun[j], tm);
        alpha[j] = __expf(mrun[j] - nm);
        mrun[j] = nm;
        float pj = __expf(s[j] - nm);
        lrun[j] = lrun[j] * alpha[j] + redsum16(pj);
        pbuf[wv][j + 8 * hl][l16] = pj;   // same-wave LDS, DS in-order
      }
#pragma unroll
      for (int nt = 0; nt < 4; ++nt)
#pragma unroll
        for (int j = 0; j < 8; ++j) accO[nt][j] *= alpha[j];
      // O += P @ V : P (16x16) as A-fragments, V as B
#pragma unroll
      for (int kk = 0; kk < 4; ++kk) {
        v2f ap;
        ap.x = pbuf[wv][l16][kk * 4 + 2 * hl];
        ap.y = pbuf[wv][l16][kk * 4 + 2 * hl + 1];
#pragma unroll
        for (int nt = 0; nt < 4; ++nt) {
          v2f bv;
          bv.x = vT[sub * 16 + kk * 4 + 2 * hl][nt * 16 + l16];
          bv.y = vT[sub * 16 + kk * 4 + 2 * hl + 1][nt * 16 + l16];
          accO[nt] = WMMA_F32(ap, bv, accO[nt]);
        }
      }
    }
  }
  // normalize and store as [B][S][H*64] (head-concat layout)
#pragma unroll
  for (int j = 0; j < 8; ++j) lrun[j] = 1.0f / lrun[j];
#pragma unroll
  for (int nt = 0; nt < 4; ++nt) {
#pragma unroll
    for (int j = 0; j < 8; ++j) {
      int srow = qt * 64 + wv * 16 + j + 8 * hl;
      Z[((size_t)b * S_LEN + srow) * 1024 + h * 64 + nt * 16 + l16] =
          accO[nt][j] * lrun[j];
    }
  }
}

// ---------------------------------------------------------------------------
// out[row] = LayerNorm(A[row] + R[row]) * g + b   (D = 1024, 256 thr/row)
// ---------------------------------------------------------------------------
__global__ __launch_bounds__(256)
void ln_residual(const float* __restrict__ A, const float* __restrict__ R,
                 const float* __restrict__ g, const float* __restrict__ be,
                 float* __restrict__ out)
{
  const size_t base = (size_t)blockIdx.x * 1024;
  __shared__ float s1[8], s2[8];
  float t[4], sum = 0.0f, sq = 0.0f;
#pragma unroll
  for (int i = 0; i < 4; ++i) {
    int c = threadIdx.x + i * 256;
    t[i] = A[base + c] + R[base + c];
    sum += t[i]; sq += t[i] * t[i];
  }
#pragma unroll
  for (int m = 1; m < 32; m <<= 1) {
    sum += __shfl_xor(sum, m, 32);
    sq  += __shfl_xor(sq,  m, 32);
  }
  if ((threadIdx.x & 31) == 0) {
    s1[threadIdx.x >> 5] = sum; s2[threadIdx.x >> 5] = sq;
  }
  __syncthreads();
  float S = 0.0f, Q2 = 0.0f;
#pragma unroll
  for (int i = 0; i < 8; ++i) { S += s1[i]; Q2 += s2[i]; }
  float mu  = S * (1.0f / 1024.0f);
  float var = Q2 * (1.0f / 1024.0f) - mu * mu;
  float inv = rsqrtf(var + 1e-5f);
#pragma unroll
  for (int i = 0; i < 4; ++i) {
    int c = threadIdx.x + i * 256;
    out[base + c] = (t[i] - mu) * inv * g[c] + be[c];
  }
}

// ---------------------------------------------------------------------------
extern "C" void kernel_launch(void* const* d_in, const int* in_sizes, int n_in,
                              void* d_out, int out_size, void* d_ws, size_t ws_size,
                              hipStream_t stream)
{
  (void)in_sizes; (void)n_in; (void)out_size; (void)ws_size;
  const float* x   = (const float*)d_in[0];
  const float* Wq  = (const float*)d_in[1];
  const float* Wk  = (const float*)d_in[2];
  const float* Wv  = (const float*)d_in[3];
  const float* Wo  = (const float*)d_in[4];
  const float* g1  = (const float*)d_in[5];
  const float* bn1 = (const float*)d_in[6];
  const float* g2  = (const float*)d_in[7];
  const float* bn2 = (const float*)d_in[8];
  const float* W1  = (const float*)d_in[9];
  const float* bb1 = (const float*)d_in[10];
  const float* W2  = (const float*)d_in[11];
  const float* bb2 = (const float*)d_in[12];
  float* out = (float*)d_out;
  float* ws  = (float*)d_ws;

  const long long M = 8192;                // B*S rows
  const size_t SEG = 8ull * 1024 * 1024;   // 8Mi floats = 32 MB
  float* q  = ws;                          // [H][M][64]
  float* k  = ws + 1 * SEG;
  float* v  = ws + 2 * SEG;
  float* z  = ws + 3 * SEG;                // [M][1024] head-concat
  float* pj = ws + 4 * SEG;                // z @ Wo
  float* y1 = ws + 5 * SEG;                // LN1 output
  float* hf = ws + 6 * SEG;                // [M][4096] (4 SEG)
  float* ff = ws + 10 * SEG;               // FFN output

  // QKV projections: per-head GEMM over z-grid (h = blockIdx.z)
  dim3 gQKV(64, 1, 16);
  gemm_wmma<<<gQKV, 128, 0, stream>>>(x, 1024, Wq, 64, 65536LL, q, 64, M * 64, 1024, nullptr, 0);
  gemm_wmma<<<gQKV, 128, 0, stream>>>(x, 1024, Wk, 64, 65536LL, k, 64, M * 64, 1024, nullptr, 0);
  gemm_wmma<<<gQKV, 128, 0, stream>>>(x, 1024, Wv, 64, 65536LL, v, 64, M * 64, 1024, nullptr, 0);

  // Flash attention: (32 q-tiles) x (16 heads) x (4 batches)
  attn_flash<<<dim3(32, 16, 4), 128, 0, stream>>>(q, k, v, z);

  // Output projection, residual + LN1
  gemm_wmma<<<dim3(64, 16, 1), 128, 0, stream>>>(z, 1024, Wo, 1024, 0, pj, 1024, 0, 1024, nullptr, 0);
  ln_residual<<<8192, 256, 0, stream>>>(pj, x, g1, bn1, y1);

  // FFN: relu(y1@W1+b1) @ W2 + b2, residual + LN2
  gemm_wmma<<<dim3(64, 64, 1), 128, 0, stream>>>(y1, 1024, W1, 4096, 0, hf, 4096, 0, 1024, bb1, 1);
  gemm_wmma<<<dim3(64, 16, 1), 128, 0, stream>>>(hf, 4096, W2, 1024, 0, ff, 1024, 0, 4096, bb2, 0);
  ln_residual<<<8192, 256, 0, stream>>>(ff, y1, g2, bn2, out);
}